// MetaNCA_54116587929662
// MI455X (gfx1250) — compile-verified
//
#include <hip/hip_runtime.h>
#include <math.h>

typedef __attribute__((ext_vector_type(2))) float v2f;
typedef __attribute__((ext_vector_type(8))) float v8f;

#define I_DIM 1024
#define J_DIM 1024
#define B_DIM 1024
#define H_DIM 32
#define HID 10
#define OUT3 33   // 1 + H

// ---------- reductions ----------------------------------------------------
__global__ void k_colsum_w(const float* __restrict__ w, float* __restrict__ colsum) {
    int j = blockIdx.x * blockDim.x + threadIdx.x;
    if (j >= J_DIM) return;
    float s = 0.f;
    for (int i = 0; i < I_DIM; ++i) s += w[(size_t)i * J_DIM + j];
    colsum[j] = s;
}

__global__ void k_rowsum_w(const float* __restrict__ w, float* __restrict__ rowsum) {
    __shared__ float sm[256];
    int i = blockIdx.x, t = threadIdx.x;
    float s = 0.f;
    for (int j = t; j < J_DIM; j += 256) s += w[(size_t)i * J_DIM + j];
    sm[t] = s; __syncthreads();
    for (int off = 128; off > 0; off >>= 1) {
        if (t < off) sm[t] += sm[t + off];
        __syncthreads();
    }
    if (t == 0) rowsum[i] = sm[0];
}

__global__ void k_colsum_h(const float* __restrict__ h, float* __restrict__ colsum) {
    int c = blockIdx.x * blockDim.x + threadIdx.x;   // c = j*H + hh
    if (c >= J_DIM * H_DIM) return;
    float s = 0.f;
    for (int i = 0; i < I_DIM; ++i) s += h[(size_t)i * (J_DIM * H_DIM) + c];
    colsum[c] = s;
}

__global__ void k_rowsum_h(const float* __restrict__ h, float* __restrict__ rowsum) {
    __shared__ float sm[256];
    int i = blockIdx.x, t = threadIdx.x;
    int hh = t & 31, g = t >> 5;                     // 8 j-groups x 32 channels
    const float* base = h + (size_t)i * (J_DIM * H_DIM);
    float s = 0.f;
    for (int j = g; j < J_DIM; j += 8) s += base[j * H_DIM + hh];
    sm[t] = s; __syncthreads();
    if (g == 0) {
        float acc = 0.f;
        for (int gg = 0; gg < 8; ++gg) acc += sm[gg * 32 + hh];
        rowsum[i * H_DIM + hh] = acc;
    }
}

// ---------- per-cell MLP update -------------------------------------------
__global__ void k_cell(const float* __restrict__ weight, const float* __restrict__ hidden,
                       const float* __restrict__ colsum_w, const float* __restrict__ rowsum_w,
                       const float* __restrict__ colsum_h, const float* __restrict__ rowsum_h,
                       const float* __restrict__ W1, const float* __restrict__ b1,
                       const float* __restrict__ W2, const float* __restrict__ b2,
                       const float* __restrict__ W3, const float* __restrict__ b3,
                       float* __restrict__ new_w, float* __restrict__ new_h) {
    __shared__ float sW1[99 * HID];      // 990
    __shared__ float sW2[HID * HID];     // 100
    __shared__ float sW3[HID * OUT3];    // 330
    __shared__ float sb1[HID], sb2[HID], sb3[OUT3];
    int t = threadIdx.x;
    for (int idx = t; idx < 99 * HID;  idx += 256) sW1[idx] = W1[idx];
    for (int idx = t; idx < HID * HID; idx += 256) sW2[idx] = W2[idx];
    for (int idx = t; idx < HID * OUT3; idx += 256) sW3[idx] = W3[idx];
    if (t < HID)  { sb1[t] = b1[t]; sb2[t] = b2[t]; }
    if (t < OUT3) sb3[t] = b3[t];
    __syncthreads();

    int c = blockIdx.x * 256 + t;        // c = i*J + j
    int i = c >> 10;
    int j = c & (J_DIM - 1);
    const float invI1 = 1.0f / (float)(I_DIM - 1);
    const float invJ1 = 1.0f / (float)(J_DIM - 1);

    float w  = weight[c];
    float fw = (colsum_w[j] - w) * invI1;
    float bw = (rowsum_w[i] - w) * invJ1;

    float a1[HID];
#pragma unroll
    for (int p = 0; p < HID; ++p)
        a1[p] = sb1[p] + w * sW1[0 * HID + p] + fw * sW1[1 * HID + p] + bw * sW1[2 * HID + p];

    float hreg[H_DIM];
    const float* hb = hidden + (size_t)c * H_DIM;
    for (int k = 0; k < H_DIM; ++k) {
        float hv = hb[k];
        hreg[k] = hv;
        float fh = (colsum_h[j * H_DIM + k] - hv) * invI1;
        float bh = (rowsum_h[i * H_DIM + k] - hv) * invJ1;
#pragma unroll
        for (int p = 0; p < HID; ++p)
            a1[p] += hv * sW1[(3 + k) * HID + p]
                   + fh * sW1[(35 + k) * HID + p]
                   + bh * sW1[(67 + k) * HID + p];
    }
#pragma unroll
    for (int p = 0; p < HID; ++p) a1[p] = fmaxf(a1[p], 0.f);   // relu 1

    float a2[HID];
#pragma unroll
    for (int q = 0; q < HID; ++q) {
        float s = sb2[q];
#pragma unroll
        for (int p = 0; p < HID; ++p) s += a1[p] * sW2[p * HID + q];
        a2[q] = fmaxf(s, 0.f);                                  // relu 2
    }

    float u0 = sb3[0];
#pragma unroll
    for (int p = 0; p < HID; ++p) u0 += a2[p] * sW3[p * OUT3 + 0];
    new_w[c] = w + u0;

    float* ob = new_h + (size_t)c * H_DIM;
    for (int k = 0; k < H_DIM; ++k) {
        float u = sb3[1 + k];
#pragma unroll
        for (int p = 0; p < HID; ++p) u += a2[p] * sW3[p * OUT3 + 1 + k];
        ob[k] = hreg[k] + u;
    }
}

// ---------- f32 WMMA GEMM: S = X (BxI) @ new_weight (IxJ) -----------------
// One wave per 16x16 output tile; K-loop of v_wmma_f32_16x16x4_f32.
// A layout (16x4 f32): lane m=lane&15 holds row m; v0/v1 = K = 2*(lane>>4)+{0,1}
// B layout (4x16 f32): lane holds col n=lane&15; v0/v1 = K rows 2*(lane>>4)+{0,1}
// C/D (16x16 f32): vgpr r -> M = r + 8*(lane>>4), N = lane&15
__global__ void k_gemm(const float* __restrict__ X, const float* __restrict__ Wn,
                       float* __restrict__ S) {
    int wave = (blockIdx.x * blockDim.x + threadIdx.x) >> 5;
    int lane = threadIdx.x & 31;
    int m  = lane & 15;
    int hi = lane >> 4;
    int row_base = (wave >> 6) << 4;     // 64 tiles over B rows
    int col_base = (wave & 63) << 4;     // 64 tiles over J cols

    const float* xrow = X + (size_t)(row_base + m) * I_DIM;
    v8f c = {};
    for (int k = 0; k < I_DIM; k += 4) {
        int ka = k + 2 * hi;
        v2f a = *(const v2f*)(xrow + ka);                 // 8B-aligned (ka even)
        v2f b;
        b.x = Wn[(size_t)ka       * J_DIM + col_base + m];
        b.y = Wn[(size_t)(ka + 1) * J_DIM + col_base + m];
        c = __builtin_amdgcn_wmma_f32_16x16x4_f32(
                /*neg_a=*/false, a, /*neg_b=*/false, b,
                /*c_mod=*/(short)0, c, /*reuse_a=*/false, /*reuse_b=*/false);
    }
#pragma unroll
    for (int r = 0; r < 8; ++r)
        S[(size_t)(row_base + r + 8 * hi) * J_DIM + col_base + m] = c[r];
}

// ---------- row softmax (in place) ----------------------------------------
__global__ void k_softmax(float* __restrict__ S) {
    __shared__ float sm[256];
    int row = blockIdx.x, t = threadIdx.x;
    float* r = S + (size_t)row * J_DIM;
    float mx = -INFINITY;
    for (int j = t; j < J_DIM; j += 256) mx = fmaxf(mx, r[j]);
    sm[t] = mx; __syncthreads();
    for (int off = 128; off > 0; off >>= 1) {
        if (t < off) sm[t] = fmaxf(sm[t], sm[t + off]);
        __syncthreads();
    }
    mx = sm[0]; __syncthreads();
    float s = 0.f;
    for (int j = t; j < J_DIM; j += 256) { float e = __expf(r[j] - mx); r[j] = e; s += e; }
    sm[t] = s; __syncthreads();
    for (int off = 128; off > 0; off >>= 1) {
        if (t < off) sm[t] += sm[t + off];
        __syncthreads();
    }
    float inv = 1.0f / sm[0];
    for (int j = t; j < J_DIM; j += 256) r[j] *= inv;
}

// ---------- launch --------------------------------------------------------
extern "C" void kernel_launch(void* const* d_in, const int* in_sizes, int n_in,
                              void* d_out, int out_size, void* d_ws, size_t ws_size,
                              hipStream_t stream) {
    (void)in_sizes; (void)n_in; (void)out_size; (void)ws_size;
    const float* X      = (const float*)d_in[0];
    const float* weight = (const float*)d_in[1];
    const float* hidden = (const float*)d_in[2];
    const float* W1 = (const float*)d_in[3];
    const float* b1 = (const float*)d_in[4];
    const float* W2 = (const float*)d_in[5];
    const float* b2 = (const float*)d_in[6];
    const float* W3 = (const float*)d_in[7];
    const float* b3 = (const float*)d_in[8];

    float* out   = (float*)d_out;
    float* probs = out;                                   // (B,J)   1M
    float* new_w = out + (size_t)B_DIM * J_DIM;           // (I,J)   1M
    float* new_h = out + 2 * (size_t)B_DIM * J_DIM;       // (I,J,H) 32M

    float* ws       = (float*)d_ws;                       // 270 KB used
    float* colsum_w = ws;                                 // J
    float* rowsum_w = ws + J_DIM;                         // I
    float* colsum_h = ws + J_DIM + I_DIM;                 // J*H
    float* rowsum_h = colsum_h + (size_t)J_DIM * H_DIM;   // I*H

    k_colsum_w<<<(J_DIM + 255) / 256, 256, 0, stream>>>(weight, colsum_w);
    k_rowsum_w<<<I_DIM, 256, 0, stream>>>(weight, rowsum_w);
    k_colsum_h<<<(J_DIM * H_DIM + 255) / 256, 256, 0, stream>>>(hidden, colsum_h);
    k_rowsum_h<<<I_DIM, 256, 0, stream>>>(hidden, rowsum_h);

    k_cell<<<(I_DIM * J_DIM) / 256, 256, 0, stream>>>(
        weight, hidden, colsum_w, rowsum_w, colsum_h, rowsum_h,
        W1, b1, W2, b2, W3, b3, new_w, new_h);

    // 64x64 = 4096 tiles, 8 waves (256 thr) per block -> 512 blocks
    k_gemm<<<(B_DIM / 16) * (J_DIM / 16) / 8, 256, 0, stream>>>(X, new_w, probs);
    k_softmax<<<B_DIM, 256, 0, stream>>>(probs);
}